// PointNet_SA_3032246911226
// MI455X (gfx1250) — compile-verified
//
#include <hip/hip_runtime.h>

// ---------------------------------------------------------------------------
// PointNet++ Set Abstraction for MI455X (gfx1250, wave32).
//  K1: farthest point sampling (serial chain, 1 block per batch, LDS-resident xyz)
//  K2: ball query (1 wave32 per centroid, ballot/popc ordered append)
//  K3: fused gather + 3x GEMM (V_WMMA_F32_16X16X4_F32) + BN/ReLU + sample-max
// ---------------------------------------------------------------------------

typedef __attribute__((ext_vector_type(2))) float v2f;
typedef __attribute__((ext_vector_type(8))) float v8f;

#define NPT 1024
#define NA  4096
#define NS  32
#define RAD2 0.04f                 // 0.2^2
#define BN_INV 0.9995003746f       // 1/sqrt(1.001)

// ------------------------------ K1: FPS ------------------------------------
__global__ __launch_bounds__(256) void fps_kernel(const float* __restrict__ xyz,
                                                  float* __restrict__ new_xyz)
{
  const int b = blockIdx.x;
  const int t = threadIdx.x;
  const int lane = t & 31, wv = t >> 5;
  extern __shared__ char smem_raw[];
  float* xs    = (float*)smem_raw;     // 4096
  float* ys    = xs + NA;              // 4096
  float* zs    = ys + NA;              // 4096
  float* redv  = zs + NA;              // 8
  int*   redi  = (int*)(redv + 8);     // 8
  int*   bcast = redi + 8;             // 1

  const float* bx = xyz + (size_t)b * NA * 3;
  for (int i = t; i < NA; i += 256) {
    xs[i] = bx[i * 3 + 0];
    ys[i] = bx[i * 3 + 1];
    zs[i] = bx[i * 3 + 2];
  }
  __syncthreads();

  float dloc[16];
#pragma unroll
  for (int k = 0; k < 16; ++k) dloc[k] = 1e10f;

  int farthest = 0;
  for (int it = 0; it < NPT; ++it) {
    const float cx = xs[farthest], cy = ys[farthest], cz = zs[farthest];
    if (t == 0) {
      float* o = new_xyz + ((size_t)b * NPT + it) * 3;
      o[0] = cx; o[1] = cy; o[2] = cz;
    }
    float bv = -1.0f;
    int   bi = 0x7fffffff;
#pragma unroll
    for (int k = 0; k < 16; ++k) {
      const int j = t + (k << 8);
      const float dx = xs[j] - cx, dy = ys[j] - cy, dz = zs[j] - cz;
      const float d  = dx * dx + dy * dy + dz * dz;
      const float dn = fminf(dloc[k], d);
      dloc[k] = dn;
      if (dn > bv) { bv = dn; bi = j; }   // j ascending -> keeps first on tie
    }
    // wave32 argmax reduction (tie -> lowest index, matches jnp.argmax)
#pragma unroll
    for (int off = 16; off > 0; off >>= 1) {
      const float ov = __shfl_down(bv, off, 32);
      const int   oi = __shfl_down(bi, off, 32);
      if (ov > bv || (ov == bv && oi < bi)) { bv = ov; bi = oi; }
    }
    if (lane == 0) { redv[wv] = bv; redi[wv] = bi; }
    __syncthreads();
    if (t == 0) {
      float v = redv[0]; int ix = redi[0];
#pragma unroll
      for (int w = 1; w < 8; ++w)
        if (redv[w] > v || (redv[w] == v && redi[w] < ix)) { v = redv[w]; ix = redi[w]; }
      bcast[0] = ix;
    }
    __syncthreads();
    farthest = bcast[0];
  }
}

// --------------------------- K2: ball query --------------------------------
__global__ __launch_bounds__(256) void ballq_kernel(const float* __restrict__ xyz,
                                                    const float* __restrict__ new_xyz,
                                                    int* __restrict__ idx)
{
  const int blk   = blockIdx.x;          // 1024 blocks: 8 batches x 128
  const int b     = blk >> 7;
  const int qbase = (blk & 127) * 8;     // 8 queries per block, 1 per wave
  const int t     = threadIdx.x;
  const int lane  = t & 31, wv = t >> 5;
  extern __shared__ char smem_raw[];
  float* xs   = (float*)smem_raw;        // 4096
  float* ys   = xs + NA;                 // 4096
  float* zs   = ys + NA;                 // 4096
  int*   qbuf = (int*)(zs + NA);         // 8 x 32

  const float* bx = xyz + (size_t)b * NA * 3;
  for (int i = t; i < NA; i += 256) {
    xs[i] = bx[i * 3 + 0];
    ys[i] = bx[i * 3 + 1];
    zs[i] = bx[i * 3 + 2];
  }
  __syncthreads();

  const int q = qbase + wv;
  const float* nq = new_xyz + ((size_t)b * NPT + q) * 3;
  const float qx = nq[0], qy = nq[1], qz = nq[2];

  int* myq = qbuf + wv * 32;
  int count = 0;
  for (int c = 0; c < NA / 32; ++c) {
    const int j = (c << 5) + lane;
    const float dx = xs[j] - qx, dy = ys[j] - qy, dz = zs[j] - qz;
    const bool in = (dx * dx + dy * dy + dz * dz) < RAD2;
    const unsigned m = (unsigned)__ballot(in);      // wave32: low 32 bits
    if (in) {
      const int pos = count + __popc(m & ((1u << lane) - 1u));
      if (pos < NS) myq[pos] = j;
    }
    count += __popc(m);
    if (count >= NS) break;                          // wave-uniform early exit
  }
  int first = 0;
  if (count > 0) first = myq[0];
  const int val = (lane < count) ? myq[lane] : first;
  idx[((size_t)b * NPT + q) * NS + lane] = val;
}

// ------------------- K3: fused gather + MLP (WMMA) + max --------------------
__global__ __launch_bounds__(128) void mlp_kernel(
    const float* __restrict__ xyz,  const float* __restrict__ points,
    const float* __restrict__ new_xyz, const int* __restrict__ idx,
    const float* __restrict__ w0, const float* __restrict__ b0,
    const float* __restrict__ g0, const float* __restrict__ be0,
    const float* __restrict__ w1, const float* __restrict__ b1,
    const float* __restrict__ g1, const float* __restrict__ be1,
    const float* __restrict__ w2, const float* __restrict__ b2,
    const float* __restrict__ g2, const float* __restrict__ be2,
    float* __restrict__ out)
{
  const int blk   = blockIdx.x;            // 4096 = 8 batches * 512
  const int b     = blk >> 9;
  const int pbase = (blk & 511) * 2;       // 2 centroids (64 rows) per block
  const int t     = threadIdx.x;           // 128 threads = 4 waves
  const int lane  = t & 31, wv = t >> 5;
  extern __shared__ char smem_raw[];
  float* A0 = (float*)smem_raw;            // 64 x 68 (layer0 input, K padded)
  float* H  = A0 + 64 * 68;                // 64 x 64 (layer0 output)
  float* H2 = A0;                          // layer1 output reuses A0 (stride 64)
  float* sb = H + 64 * 64;                 // 128 scale
  float* tb = sb + 128;                    // 128 bias
  float* nx = tb + 128;                    // 8 (centroid xyz)
  int* ridx = (int*)(nx + 8);              // 64 gathered indices
  int* cmax = ridx + 64;                   // 2 x 128 column max (int-punned f32)

  if (t < 64) ridx[t] = idx[((size_t)b * NPT + pbase) * NS + t];
  if (t < 6)  nx[t]   = new_xyz[((size_t)b * NPT + pbase) * 3 + t];
  for (int e = t; e < 256; e += 128) cmax[e] = 0;
  __syncthreads();

  // gather: [points_feat(64) | xyz - centroid(3) | 0 pad] per row
  for (int e = t; e < 64 * 68; e += 128) {
    const int r = e / 68, c = e - r * 68;
    const int pl = r >> 5;
    const int pi = ridx[r];
    float v;
    if (c < 64)      v = points[((size_t)b * NA + pi) * 64 + c];
    else if (c < 67) v = xyz[((size_t)b * NA + pi) * 3 + (c - 64)] - nx[pl * 3 + (c - 64)];
    else             v = 0.0f;
    A0[e] = v;
  }
  if (t < 64) { const float s = g0[t] * BN_INV; sb[t] = s; tb[t] = s * b0[t] + be0[t]; }
  __syncthreads();

  const int half = lane >> 4;              // A/B K-half select (wave32 layout)
  const int nl   = lane & 15;
  const int m0   = wv << 4;                // each wave owns 16 rows

  // ---- layer 0: A0[64x67] * w0[67x64] -> BN/ReLU -> H
  for (int nt = 0; nt < 4; ++nt) {
    v8f acc = {0.f, 0.f, 0.f, 0.f, 0.f, 0.f, 0.f, 0.f};
    const int nc = (nt << 4) + nl;
    for (int kt = 0; kt < 17; ++kt) {
      const int kk = (kt << 2) + (half << 1);
      const v2f a = *(const v2f*)&A0[(m0 + nl) * 68 + kk];
      v2f bf;
      bf.x = (kk     < 67) ? w0[kk * 64 + nc]       : 0.0f;
      bf.y = (kk + 1 < 67) ? w0[(kk + 1) * 64 + nc] : 0.0f;
      acc = __builtin_amdgcn_wmma_f32_16x16x4_f32(false, a, false, bf,
                                                  (short)0, acc, false, false);
    }
    const float s = sb[nc], tt = tb[nc];
#pragma unroll
    for (int r = 0; r < 8; ++r)
      H[(m0 + (half << 3) + r) * 64 + nc] = fmaxf(acc[r] * s + tt, 0.0f);
  }
  __syncthreads();
  if (t < 64) { const float s = g1[t] * BN_INV; sb[t] = s; tb[t] = s * b1[t] + be1[t]; }
  __syncthreads();

  // ---- layer 1: H[64x64] * w1[64x64] -> BN/ReLU -> H2
  for (int nt = 0; nt < 4; ++nt) {
    v8f acc = {0.f, 0.f, 0.f, 0.f, 0.f, 0.f, 0.f, 0.f};
    const int nc = (nt << 4) + nl;
    for (int kt = 0; kt < 16; ++kt) {
      const int kk = (kt << 2) + (half << 1);
      const v2f a = *(const v2f*)&H[(m0 + nl) * 64 + kk];
      v2f bf;
      bf.x = w1[kk * 64 + nc];
      bf.y = w1[(kk + 1) * 64 + nc];
      acc = __builtin_amdgcn_wmma_f32_16x16x4_f32(false, a, false, bf,
                                                  (short)0, acc, false, false);
    }
    const float s = sb[nc], tt = tb[nc];
#pragma unroll
    for (int r = 0; r < 8; ++r)
      H2[(m0 + (half << 3) + r) * 64 + nc] = fmaxf(acc[r] * s + tt, 0.0f);
  }
  __syncthreads();
  if (t < 128) { const float s = g2[t] * BN_INV; sb[t] = s; tb[t] = s * b2[t] + be2[t]; }
  __syncthreads();

  // ---- layer 2: H2[64x64] * w2[64x128] -> BN/ReLU -> max over 32 samples
  const int pl = wv >> 1;                  // which centroid this wave's rows belong to
  for (int nt = 0; nt < 8; ++nt) {
    v8f acc = {0.f, 0.f, 0.f, 0.f, 0.f, 0.f, 0.f, 0.f};
    const int nc = (nt << 4) + nl;
    for (int kt = 0; kt < 16; ++kt) {
      const int kk = (kt << 2) + (half << 1);
      const v2f a = *(const v2f*)&H2[(m0 + nl) * 64 + kk];
      v2f bf;
      bf.x = w2[kk * 128 + nc];
      bf.y = w2[(kk + 1) * 128 + nc];
      acc = __builtin_amdgcn_wmma_f32_16x16x4_f32(false, a, false, bf,
                                                  (short)0, acc, false, false);
    }
    const float s = sb[nc], tt = tb[nc];
    float pm = 0.0f;
#pragma unroll
    for (int r = 0; r < 8; ++r) pm = fmaxf(pm, fmaxf(acc[r] * s + tt, 0.0f));
    pm = fmaxf(pm, __shfl_xor(pm, 16, 32));          // combine the two M-halves
    if (lane < 16)                                   // ReLU>=0 -> int max == float max
      atomicMax(&cmax[(pl << 7) + nc], __float_as_int(pm));
  }
  __syncthreads();
  for (int e = t; e < 256; e += 128) {
    const int p2 = e >> 7, c = e & 127;
    out[((size_t)b * NPT + pbase + p2) * 128 + c] = __int_as_float(cmax[e]);
  }
}

// ------------------------------- launcher -----------------------------------
extern "C" void kernel_launch(void* const* d_in, const int* in_sizes, int n_in,
                              void* d_out, int out_size, void* d_ws, size_t ws_size,
                              hipStream_t stream) {
  (void)in_sizes; (void)n_in; (void)out_size; (void)ws_size;
  const float* xyz    = (const float*)d_in[0];
  const float* points = (const float*)d_in[1];
  const float* w0 = (const float*)d_in[2];
  const float* b0 = (const float*)d_in[3];
  const float* g0 = (const float*)d_in[4];
  const float* be0 = (const float*)d_in[5];
  const float* w1 = (const float*)d_in[6];
  const float* b1 = (const float*)d_in[7];
  const float* g1 = (const float*)d_in[8];
  const float* be1 = (const float*)d_in[9];
  const float* w2 = (const float*)d_in[10];
  const float* b2 = (const float*)d_in[11];
  const float* g2 = (const float*)d_in[12];
  const float* be2 = (const float*)d_in[13];

  float* out        = (float*)d_out;
  float* new_xyz    = out;                    // 8*1024*3
  float* new_points = out + 8 * NPT * 3;      // 8*1024*128
  int*   idx        = (int*)d_ws;             // 8*1024*32 ints (1 MB scratch)

  const size_t smem_fps  = (size_t)(3 * NA) * 4 + 8 * 4 + 8 * 4 + 16;
  const size_t smem_ball = (size_t)(3 * NA) * 4 + 8 * 32 * 4;
  const size_t smem_mlp  = (size_t)(64 * 68 + 64 * 64 + 128 + 128 + 8) * 4
                         + (size_t)(64 + 256) * 4;

  fps_kernel  <<<8,    256, smem_fps,  stream>>>(xyz, new_xyz);
  ballq_kernel<<<1024, 256, smem_ball, stream>>>(xyz, new_xyz, idx);
  mlp_kernel  <<<4096, 128, smem_mlp,  stream>>>(xyz, points, new_xyz, idx,
                                                 w0, b0, g0, be0,
                                                 w1, b1, g1, be1,
                                                 w2, b2, g2, be2,
                                                 new_points);
}